// VoxelProjection_78245714198739
// MI455X (gfx1250) — compile-verified
//
#include <hip/hip_runtime.h>

// ---------------------------------------------------------------------------
// BEV voxel projection (gather + weight + z-reduce), tuned for MI455X:
//  - feats (176 MB) kept L2-resident (192 MB L2): regular-temporal gathers
//  - streamed-once data (indices in, BEV out) uses non-temporal hints
//  - index rows staged via CDNA5 async global->LDS ops (ASYNCcnt path)
// ---------------------------------------------------------------------------

#define Z_LEVELS 6
#define CELLS    28800        // 240*120 BEV cells
#define L_VOX    172800       // Z_LEVELS*CELLS
#define C_CH     64
#define IMG_W    448
#define HW       114688       // 256*448
#define N_CAM    6
#define TILE     128          // cells per block; 28800 % 128 == 0

#define AS1 __attribute__((address_space(1)))
#define AS3 __attribute__((address_space(3)))

#if defined(__HIP_DEVICE_COMPILE__) && \
    __has_builtin(__builtin_amdgcn_global_load_async_to_lds_b32)
#define HAVE_ASYNC_LDS 1
#else
#define HAVE_ASYNC_LDS 0
#endif

#if HAVE_ASYNC_LDS
// Builtin expects non-const int* in AS1 (global) and AS3 (LDS).
__device__ __forceinline__ AS1 int* as_global_i32(const void* p) {
    return (AS1 int*)(int*)const_cast<void*>(p);
}
__device__ __forceinline__ AS3 int* as_lds_i32(void* p) {
    return (AS3 int*)(int*)p;
}
#endif

__global__ __launch_bounds__(TILE) void VoxelProjection_78245714198739_kernel(
    const float* __restrict__ fish,     // [4, 64, 256, 448]
    const float* __restrict__ pv,       // [1, 64, 256, 448]
    const float* __restrict__ front,    // [1, 64, 256, 448]
    const int*   __restrict__ uu,       // [6, L_VOX]
    const int*   __restrict__ vv,       // [6, L_VOX]
    const int*   __restrict__ valid,    // [6, L_VOX]
    const float* __restrict__ density,  // [6, L_VOX]
    float*       __restrict__ out)      // [384, 28800]
{
    const int cam  = blockIdx.y;
    const int tid  = threadIdx.x;
    const int cell = blockIdx.x * TILE + tid;

    // Concatenated camera features: cams 0-3 = fish, 4 = pv, 5 = front.
    const float* __restrict__ feats =
        (cam < 4) ? (fish + (size_t)cam * ((size_t)C_CH * HW))
                  : ((cam == 4) ? pv : front);

    const size_t vbase = (size_t)cam * L_VOX + (size_t)cell;

    int   pix[Z_LEVELS];
    float wgt[Z_LEVELS];

#if HAVE_ASYNC_LDS
    // Stage the 6 z-rows of (uu, vv, valid, density) for this cell tile with
    // async global->LDS DMA (no VGPR round-trip, tracked by ASYNCcnt).
    __shared__ int   s_u[Z_LEVELS * TILE];
    __shared__ int   s_v[Z_LEVELS * TILE];
    __shared__ int   s_m[Z_LEVELS * TILE];
    __shared__ float s_d[Z_LEVELS * TILE];

    #pragma unroll
    for (int z = 0; z < Z_LEVELS; ++z) {
        const size_t g = vbase + (size_t)z * CELLS;
        const int    s = z * TILE + tid;
        __builtin_amdgcn_global_load_async_to_lds_b32(
            as_global_i32(uu + g),      as_lds_i32(&s_u[s]), 0, 0);
        __builtin_amdgcn_global_load_async_to_lds_b32(
            as_global_i32(vv + g),      as_lds_i32(&s_v[s]), 0, 0);
        __builtin_amdgcn_global_load_async_to_lds_b32(
            as_global_i32(valid + g),   as_lds_i32(&s_m[s]), 0, 0);
        __builtin_amdgcn_global_load_async_to_lds_b32(
            as_global_i32(density + g), as_lds_i32(&s_d[s]), 0, 0);
    }
#if __has_builtin(__builtin_amdgcn_s_wait_asynccnt)
    __builtin_amdgcn_s_wait_asynccnt(0);
#else
    asm volatile("s_wait_asynccnt 0" ::: "memory");
#endif
    // Each thread consumes exactly the LDS slots its own wave's async loads
    // wrote, so the per-wave ASYNCcnt wait is sufficient (no barrier).
    #pragma unroll
    for (int z = 0; z < Z_LEVELS; ++z) {
        const int s = z * TILE + tid;
        pix[z] = s_v[s] * IMG_W + s_u[s];
        wgt[z] = s_m[s] ? s_d[s] : 0.0f;
    }
#else
    // Fallback: straight coalesced NT loads into registers.
    #pragma unroll
    for (int z = 0; z < Z_LEVELS; ++z) {
        const size_t g = vbase + (size_t)z * CELLS;
        const int   u = __builtin_nontemporal_load(uu + g);
        const int   v = __builtin_nontemporal_load(vv + g);
        const int   m = __builtin_nontemporal_load(valid + g);
        const float d = __builtin_nontemporal_load(density + g);
        pix[z] = v * IMG_W + u;
        wgt[z] = m ? d : 0.0f;
    }
#endif

    // 64-channel loop: 6 independent L2 gathers + FMAs per channel.
    // Unroll x4 -> ~24 loads in flight per thread to cover L2 latency.
    float* __restrict__ op = out + ((size_t)cam * C_CH) * CELLS + (size_t)cell;
    #pragma unroll 4
    for (int c = 0; c < C_CH; ++c) {
        const float* __restrict__ plane = feats + (size_t)c * HW;
        float acc = 0.0f;
        #pragma unroll
        for (int z = 0; z < Z_LEVELS; ++z)
            acc = fmaf(wgt[z], plane[pix[z]], acc);
        // NT store: keep the 44 MB output stream out of L2 so the 176 MB
        // feature set stays resident.
        __builtin_nontemporal_store(acc, op + (size_t)c * CELLS);
    }
}

extern "C" void kernel_launch(void* const* d_in, const int* in_sizes, int n_in,
                              void* d_out, int out_size, void* d_ws, size_t ws_size,
                              hipStream_t stream) {
    (void)in_sizes; (void)n_in; (void)out_size; (void)d_ws; (void)ws_size;
    const float* fish    = (const float*)d_in[0];
    const float* pv      = (const float*)d_in[1];
    const float* front   = (const float*)d_in[2];
    const int*   uu      = (const int*)d_in[3];
    const int*   vv      = (const int*)d_in[4];
    const int*   valid   = (const int*)d_in[5];
    const float* density = (const float*)d_in[6];
    float*       out     = (float*)d_out;

    dim3 grid(CELLS / TILE, N_CAM);   // 225 x 6 blocks
    dim3 block(TILE);                 // 128 threads = 4 waves
    VoxelProjection_78245714198739_kernel<<<grid, block, 0, stream>>>(
        fish, pv, front, uu, vv, valid, density, out);
}